// APTEncoderWrapper_5128190951572
// MI455X (gfx1250) — compile-verified
//
#include <hip/hip_runtime.h>
#include <cstdint>

// ---- problem constants (from reference) ----
#define Bc   8
#define Hc   32
#define Wc   32
#define Tc   4
#define BSc  3
#define Dc   768
#define Pc   2560
#define HWT  (Hc * Wc * Tc)        // 4096 cells per batch
#define TOKF (BSc * Dc)            // 2304 floats per patch token
#define TOKB (TOKF * 4)            // 9216 bytes per token
#define CPG  16                    // cells per TDM group (max 16-bit gather rows)
#define GPB  (HWT / CPG)           // 256 groups per batch
#define LDSB (CPG * TOKB)          // 147456 bytes of LDS staging (<= 320KB/WG)

typedef unsigned int       u32;
typedef unsigned long long u64;
typedef u32 v4u __attribute__((ext_vector_type(4)));
typedef int v4i __attribute__((ext_vector_type(4)));
typedef int v8i __attribute__((ext_vector_type(8)));

// ---- kernel 1: init id_map to -1 ----
__global__ void apt_init_idmap(int* __restrict__ idm) {
    int i = blockIdx.x * blockDim.x + threadIdx.x;
    if (i < Bc * HWT) idm[i] = -1;
}

// ---- kernel 2: scatter patch ids into cell->patch map ----
__global__ void apt_scatter_ids(const int* __restrict__ pos, int* __restrict__ idm) {
    int i = blockIdx.x * blockDim.x + threadIdx.x;   // over B*P patches
    if (i >= Bc * Pc) return;
    int b = i / Pc;
    int p = i - b * Pc;
    int4 q = ((const int4*)pos)[i];                  // (y, x, size, t)
    int* base = idm + b * HWT;
    for (int dy = 0; dy < q.z; ++dy)
        for (int dx = 0; dx < q.z; ++dx)
            base[((q.x + dy) * Wc + (q.y + dx)) * Tc + q.w] = p;
}

__device__ __forceinline__ u32 tdm_addr_hi(u64 a) {
    // D# group0 bits 120:64 = 57-bit global addr; bits 127:126 = type(2)
    return (u32)((a >> 32) & 0x01FFFFFFull) | (2u << 30);
}

__device__ __forceinline__ u32 pack2(u32 id, int l0, int l1) {
    return ((u32)__builtin_amdgcn_readlane((int)id, l0) & 0xFFFFu) |
           (((u32)__builtin_amdgcn_readlane((int)id, l1) & 0xFFFFu) << 16);
}

// ---- kernel 3: TDM gather (global --16-row gather--> LDS --> global) ----
// One wave per group of CPG consecutive cells of one batch.
__global__ void __launch_bounds__(32)
apt_tdm_gather(const float* __restrict__ tok, const int* __restrict__ idm,
               float* __restrict__ out) {
    extern __shared__ float smem[];
    const int lane  = threadIdx.x;
    const int b     = blockIdx.x / GPB;
    const int cell0 = (blockIdx.x - b * GPB) * CPG;

    // per-lane: patch id for this cell; -1 (uncovered) -> 0xFFFF = OOB row
    // (TDM gather OOB reads return zero == reference's zero-fill semantics)
    u32 id = 0xFFFFu;
    if (lane < CPG) {
        int v = idm[b * HWT + cell0 + lane];
        id = (v < 0) ? 0xFFFFu : ((u32)v & 0xFFFFu);
    }
    // pack 16 x 16-bit row indices into wave-uniform words via v_readlane
    u32 iw0 = pack2(id, 0, 1),   iw1 = pack2(id, 2, 3);
    u32 iw2 = pack2(id, 4, 5),   iw3 = pack2(id, 6, 7);
    u32 iw4 = pack2(id, 8, 9),   iw5 = pack2(id, 10, 11);
    u32 iw6 = pack2(id, 12, 13), iw7 = pack2(id, 14, 15);

    const u32 lds_addr =
        (u32)__builtin_amdgcn_readfirstlane((int)(u32)(uintptr_t)smem);

    // ---------- load descriptor: gather 16 token rows into LDS ----------
    const u64 gsrc = (u64)(uintptr_t)tok + (u64)b * (u64)Pc * (u64)TOKB;
    v4u g0l; // count=1 | gather_index_size=16b | gather_mode=1 ; lds ; addr
    g0l.x = 0x80000001u;
    g0l.y = lds_addr;
    g0l.z = (u32)gsrc;
    g0l.w = tdm_addr_hi(gsrc);

    v8i g1l;
    g1l[0] = 0x00020000;                         // data_size = 4 bytes
    g1l[1] = (int)(((u32)TOKF & 0xFFFFu) << 16); // tensor_dim0 = 2304 (row len)
    g1l[2] = (int)((u32)Pc << 16);               // tensor_dim1 = P (row-index bound)
    g1l[3] = (int)(((u32)TOKF & 0xFFFFu) << 16); // tile_dim0 = 2304
    g1l[4] = CPG;                                // tile_dim1 = #valid indices (16)
    g1l[5] = TOKF;                               // tensor_dim0_stride = 2304
    g1l[6] = 0;
    g1l[7] = 0;

    v4i g2l; g2l.x = (int)iw0; g2l.y = (int)iw1; g2l.z = (int)iw2; g2l.w = (int)iw3;
    v4i g3l; g3l.x = (int)iw4; g3l.y = (int)iw5; g3l.z = (int)iw6; g3l.w = (int)iw7;
    v4i g3z = {0, 0, 0, 0};
    v8i g5z = {0, 0, 0, 0, 0, 0, 0, 0};
    (void)g5z;

    // ---------- store descriptor: contiguous 16-cell span to output ----------
    const u64 gdst = (u64)(uintptr_t)out +
                     ((u64)b * (u64)HWT + (u64)cell0) * (u64)TOKB;
    const u32 NEL = (u32)(CPG * TOKF);           // 36864 elements (fits 16-bit tile_dim0)
    v4u g0s;
    g0s.x = 0x00000001u;                         // count=1, normal mode
    g0s.y = lds_addr;
    g0s.z = (u32)gdst;
    g0s.w = tdm_addr_hi(gdst);

    v8i g1s;
    g1s[0] = 0x00020000;                         // data_size = 4 bytes
    g1s[1] = (int)((NEL & 0xFFFFu) << 16);       // tensor_dim0 lo16
    g1s[2] = (int)(((NEL >> 16) & 0xFFFFu) | (1u << 16)); // dim0 hi | tensor_dim1=1
    g1s[3] = (int)((NEL & 0xFFFFu) << 16);       // tile_dim0 = 36864
    g1s[4] = 0;                                  // tile_dim1/2 unused
    g1s[5] = (int)NEL;                           // tensor_dim0_stride
    g1s[6] = 0;
    g1s[7] = 0;

#if __clang_major__ >= 23
    __builtin_amdgcn_tensor_load_to_lds(g0l, g1l, g2l, g3l, g5z, 0);
    __builtin_amdgcn_s_wait_tensorcnt(0);
    __builtin_amdgcn_tensor_store_from_lds(g0s, g1s, g3z, g3z, g5z, 0);
#else
    __builtin_amdgcn_tensor_load_to_lds(g0l, g1l, g2l, g3l, 0);
    __builtin_amdgcn_s_wait_tensorcnt(0);
    __builtin_amdgcn_tensor_store_from_lds(g0s, g1s, g3z, g3z, 0);
#endif
    // s_endpgm performs an implicit wait-idle (covers the store's TENSORcnt)
}

extern "C" void kernel_launch(void* const* d_in, const int* in_sizes, int n_in,
                              void* d_out, int out_size, void* d_ws, size_t ws_size,
                              hipStream_t stream) {
    (void)in_sizes; (void)n_in; (void)out_size; (void)ws_size;
    const float* tok = (const float*)d_in[0];   // modality_tokens [B, P*BS, D] f32
    const int*   pos = (const int*)d_in[1];     // positions       [B, P, 4]  i32
    float*       out = (float*)d_out;           // [B,H,W,T,BS,D] f32
    int*         idm = (int*)d_ws;              // cell -> patch id, B*HWT ints

    apt_init_idmap  <<<(Bc * HWT + 255) / 256, 256, 0, stream>>>(idm);
    apt_scatter_ids <<<(Bc * Pc  + 255) / 256, 256, 0, stream>>>(pos, idm);
    apt_tdm_gather  <<<Bc * GPB, 32, LDSB, stream>>>(tok, idm, out);
}